// Speller_16200616640572
// MI455X (gfx1250) — compile-verified
//
#include <hip/hip_runtime.h>
#include <math.h>

typedef float v2f __attribute__((ext_vector_type(2)));
typedef float v8f __attribute__((ext_vector_type(8)));
typedef unsigned int u32x4 __attribute__((ext_vector_type(4)));
typedef int i32x4 __attribute__((ext_vector_type(4)));
typedef int i32x8 __attribute__((ext_vector_type(8)));

#define Bsz 64
#define Tlen 1500
#define Dd 512
#define Hh 256
#define Vv 32
#define Ll 128
#define NEGF (-1e30f)
#define CHUNK 96
#define NP 16

__device__ __forceinline__ float sigmf_(float x) { return 1.0f / (1.0f + __expf(-x)); }

// ---------------------------------------------------------------------------
// Generic f32 WMMA GEMM:  C(64 x 16*tilesN) = A(64 x K) * Bt(16*tilesN x K)^T
// A row-major w/ stride lda, Bt row-major (N x K) w/ stride ldb (weight layout).
// One 16x16 tile per wave using V_WMMA_F32_16X16X4_F32, dual accumulators to
// break the D->C dependency chain.
// ---------------------------------------------------------------------------
__global__ __launch_bounds__(128) void gemm_wmma_f32(
    const float* __restrict__ A, int lda,
    const float* __restrict__ Bt, int ldb,
    const float* __restrict__ bias,
    float* __restrict__ C, int ldc,
    int K, int tilesN) {
  int wid  = blockIdx.x * 4 + (threadIdx.x >> 5);
  int lane = threadIdx.x & 31;
  int tileM = wid & 3;        // M = 64 -> 4 tiles always
  int tileN = wid >> 2;
  if (tileN >= tilesN) return;

  int r  = lane & 15;
  int kh = (lane >> 4) << 1;  // 0 or 2 (k-pair held by this half-wave)
  const float* arow = A  + (size_t)(tileM * 16 + r) * lda;
  const float* brow = Bt + (size_t)(tileN * 16 + r) * ldb;

  v8f acc0 = {}, acc1 = {};
#pragma unroll 4
  for (int k0 = 0; k0 < K; k0 += 8) {   // K is a multiple of 8 everywhere here
    v2f a0 = *(const v2f*)(arow + k0 + kh);
    v2f b0 = *(const v2f*)(brow + k0 + kh);
    acc0 = __builtin_amdgcn_wmma_f32_16x16x4_f32(
        false, a0, false, b0, (short)0, acc0, false, false);
    v2f a1 = *(const v2f*)(arow + k0 + 4 + kh);
    v2f b1 = *(const v2f*)(brow + k0 + 4 + kh);
    acc1 = __builtin_amdgcn_wmma_f32_16x16x4_f32(
        false, a1, false, b1, (short)0, acc1, false, false);
  }
  v8f acc = acc0 + acc1;

  int n = tileN * 16 + (lane & 15);
  int mbase = tileM * 16 + ((lane >> 4) << 3);  // lanes 16-31 hold M=8..15
  float bv = bias ? bias[n] : 0.0f;
#pragma unroll
  for (int i = 0; i < 8; ++i)
    C[(size_t)(mbase + i) * ldc + n] = acc[i] + bv;
}

// ---------------------------------------------------------------------------
// One-time prep: Wcat[n][k] = k<512 ? W_ih[n][32+k] : W_hh[n][k-512];
// bb[n] = b_ih[n] + b_hh[n]
// ---------------------------------------------------------------------------
__global__ void prep_wcat(const float* __restrict__ W_ih,
                          const float* __restrict__ W_hh,
                          const float* __restrict__ b_ih,
                          const float* __restrict__ b_hh,
                          float* __restrict__ Wcat, float* __restrict__ bb) {
  int idx = blockIdx.x * 256 + threadIdx.x;  // 2048*1024
  int n = idx >> 10, k = idx & 1023;
  Wcat[idx] = (k < Dd) ? W_ih[(size_t)n * (Vv + Dd) + Vv + k]
                       : W_hh[(size_t)n * Dd + (k - Dd)];
  if (k == 0) bb[n] = b_ih[n] + b_hh[n];
}

__global__ void prep_psiT(const float* __restrict__ psi_w, float* __restrict__ psiT) {
  int idx = blockIdx.x * 256 + threadIdx.x;  // 512*256, psiT[d][h] = psi_w[h][d]
  int d = idx >> 8, h = idx & 255;
  psiT[idx] = psi_w[(size_t)h * Dd + d];
}

// ch[b][0:512]=ctx=x[b,0,:], ch[b][512:1024]=h=0, c=0
__global__ void init_state(const float* __restrict__ x,
                           float* __restrict__ ch, float* __restrict__ c) {
  int idx = blockIdx.x * 256 + threadIdx.x;  // 64*512
  int b = idx >> 9, d = idx & 511;
  ch[(size_t)b * 1024 + d]       = x[(size_t)b * Tlen * Dd + d];
  ch[(size_t)b * 1024 + 512 + d] = 0.0f;
  c[idx] = 0.0f;
}

// ---------------------------------------------------------------------------
// LSTM cell: add biases + one-hot token column of W_ih, apply gates.
// ---------------------------------------------------------------------------
__global__ void lstm_cell(const float* __restrict__ gates,
                          const float* __restrict__ bb,
                          const float* __restrict__ W_ih,
                          const int* __restrict__ labels,
                          float* __restrict__ ch, float* __restrict__ c,
                          int step) {
  int idx = blockIdx.x * 256 + threadIdx.x;  // 64*512
  int b = idx >> 9, j = idx & 511;
  int tok = 1;  // SOS
  if (step > 0) {
    int lb = labels[b * Ll + step - 1];
    tok = (lb == -100) ? 0 : lb;
  }
  const float* g = gates + (size_t)b * 2048;
  float iv = g[j]          + bb[j]          + W_ih[(size_t)j * 544 + tok];
  float fv = g[512 + j]    + bb[512 + j]    + W_ih[(size_t)(512 + j) * 544 + tok];
  float gv = g[1024 + j]   + bb[1024 + j]   + W_ih[(size_t)(1024 + j) * 544 + tok];
  float ov = g[1536 + j]   + bb[1536 + j]   + W_ih[(size_t)(1536 + j) * 544 + tok];
  float cn = sigmf_(fv) * c[idx] + sigmf_(iv) * tanhf(gv);
  c[idx] = cn;
  ch[(size_t)b * 1024 + 512 + j] = sigmf_(ov) * tanhf(cn);
}

// ---------------------------------------------------------------------------
// Fused attention partial: one block per (batch, T-partition).
// Stage: TDM tensor_load_to_lds DMAs the 96x512 f32 tile into LDS (OOB rows
//        auto-zero-filled via tensor_dim1), tracked by TENSORcnt.
// Pass1: scores = x.q from LDS (wave dot + shfl reduce), block max.
// Pass2: e = exp(s-m), partial context from LDS tile.
// ---------------------------------------------------------------------------
__global__ __launch_bounds__(256) void attn_partial(
    const float* __restrict__ x, const float* __restrict__ q,
    const int* __restrict__ seqlen,
    float* __restrict__ pm, float* __restrict__ pl, float* __restrict__ pctx) {
  extern __shared__ float sm[];
  float* xs  = sm;                 // CHUNK*512 staged x tile
  float* sc  = sm + CHUNK * Dd;    // CHUNK scores -> exp weights
  float* red = sc + CHUNK;         // reductions

  int blk = blockIdx.x;
  int b = blk >> 4, p = blk & (NP - 1);
  int tid = threadIdx.x, wave = tid >> 5, lane = tid & 31;
  int len = seqlen[b];
  int t0 = p * CHUNK;
  int rows_avail = Tlen - t0;      // TDM zero-fills tile rows >= this

#if defined(__gfx1250__) && __has_builtin(__builtin_amdgcn_tensor_load_to_lds)
  if (wave == 0) {                 // wave-uniform: one TDM op per workgroup
    const float* gsrc = x + (size_t)b * Tlen * Dd + (size_t)t0 * Dd;
    unsigned long long ga = (unsigned long long)(uintptr_t)gsrc;
    unsigned lds_base = (unsigned)(uintptr_t)xs;  // generic LDS ptr low 32 = LDS offset
    // D# group 0: count=1(valid), lds_addr, global_addr[56:0], type=2
    u32x4 g0 = { 1u,
                 lds_base,
                 (unsigned)(ga & 0xffffffffu),
                 (unsigned)((ga >> 32) & 0x01ffffffu) | (2u << 30) };
    // D# group 1: data_size=4B(code 2); tensor_dim0=512; tensor_dim1=rows_avail;
    // tile_dim0=512; tile_dim1=CHUNK; tensor_dim0_stride=512
    i32x8 g1 = { (int)(2u << 16),
                 (int)(512u << 16),
                 (int)(((unsigned)rows_avail & 0xffffu) << 16),
                 (int)((((unsigned)rows_avail >> 16) & 0xffffu) | (512u << 16)),
                 (int)CHUNK,
                 512,
                 0, 0 };
    i32x4 gz4 = { 0, 0, 0, 0 };    // groups 2/3 unused (2D tensor)
    i32x8 gz8 = { 0, 0, 0, 0, 0, 0, 0, 0 };
    // 6-arg form (amdgpu-toolchain clang-23)
    __builtin_amdgcn_tensor_load_to_lds(g0, g1, gz4, gz4, gz8, 0);
  }
#endif

  // overlap: load q while the DMA runs
  float4 qr[4];
  const float* qb = q + (size_t)b * Dd;
#pragma unroll
  for (int it = 0; it < 4; ++it)
    qr[it] = *(const float4*)(qb + it * 128 + lane * 4);

#if defined(__gfx1250__) && __has_builtin(__builtin_amdgcn_tensor_load_to_lds)
  if (wave == 0) __builtin_amdgcn_s_wait_tensorcnt(0);
#else
  // fallback: explicit staged copy global -> LDS
  for (int tl = wave; tl < CHUNK; tl += 8) {
    int t = t0 + tl;
    if (t < Tlen) {
      const float* xr = x + ((size_t)b * Tlen + t) * Dd;
      __builtin_prefetch(xr + 8 * Dd, 0, 0);
#pragma unroll
      for (int it = 0; it < 4; ++it)
        *(float4*)(xs + tl * Dd + it * 128 + lane * 4) =
            *(const float4*)(xr + it * 128 + lane * 4);
    } else {
#pragma unroll
      for (int it = 0; it < 4; ++it)
        *(float4*)(xs + tl * Dd + it * 128 + lane * 4) = make_float4(0, 0, 0, 0);
    }
  }
#endif
  __syncthreads();  // tile visible to all waves

  // pass 1: scores from LDS
  float wmax = NEGF;
  for (int tl = wave; tl < CHUNK; tl += 8) {
    int t = t0 + tl;
    float dot = 0.0f;
#pragma unroll
    for (int it = 0; it < 4; ++it) {
      float4 xv = *(const float4*)(xs + tl * Dd + it * 128 + lane * 4);
      dot += xv.x * qr[it].x + xv.y * qr[it].y + xv.z * qr[it].z + xv.w * qr[it].w;
    }
#pragma unroll
    for (int off = 16; off; off >>= 1) dot += __shfl_xor(dot, off, 32);
    if (lane == 0) {
      float s = (t < len) ? dot : NEGF;
      sc[tl] = s;
      wmax = fmaxf(wmax, s);
    }
  }
  if (lane == 0) red[wave] = wmax;
  __syncthreads();
  if (tid == 0) {
    float m = red[0];
    for (int w = 1; w < 8; ++w) m = fmaxf(m, red[w]);
    red[8] = m;
  }
  __syncthreads();
  float m = red[8];
  if (tid < CHUNK) {
    float s = sc[tid];
    sc[tid] = (s <= 0.5f * NEGF) ? 0.0f : __expf(s - m);  // fully-masked guard
  }
  __syncthreads();

  // pass 2: partial context
  float c0 = 0.0f, c1 = 0.0f;
  int d0 = tid, d1 = tid + 256;  // stride-1 per wave -> bank-conflict free
  for (int t = 0; t < CHUNK; ++t) {
    float w = sc[t];
    c0 += w * xs[t * Dd + d0];
    c1 += w * xs[t * Dd + d1];
  }
  if (tid == 0) {
    float l = 0.0f;
    for (int t = 0; t < CHUNK; ++t) l += sc[t];
    pm[blk] = m;
    pl[blk] = l;
  }
  float* pc = pctx + (size_t)blk * Dd;
  pc[d0] = c0;
  pc[d1] = c1;
}

// Log-sum-exp merge of NP partials -> normalized context into ch[:, 0:512]
__global__ void attn_combine(const float* __restrict__ pm,
                             const float* __restrict__ pl,
                             const float* __restrict__ pctx,
                             float* __restrict__ ch) {
  int b = blockIdx.x, tid = threadIdx.x;
  float M = NEGF;
#pragma unroll
  for (int p = 0; p < NP; ++p) M = fmaxf(M, pm[b * NP + p]);
  float Lsum = 0.0f, c0 = 0.0f, c1 = 0.0f;
#pragma unroll
  for (int p = 0; p < NP; ++p) {
    float s = pm[b * NP + p];
    float w = (s <= 0.5f * NEGF) ? 0.0f : __expf(s - M);
    Lsum += pl[b * NP + p] * w;
    const float* pc = pctx + (size_t)(b * NP + p) * Dd;
    c0 += w * pc[tid];
    c1 += w * pc[tid + 256];
  }
  float inv = 1.0f / Lsum;
  ch[(size_t)b * 1024 + tid]       = c0 * inv;
  ch[(size_t)b * 1024 + tid + 256] = c1 * inv;
}

// ---------------------------------------------------------------------------
extern "C" void kernel_launch(void* const* d_in, const int* in_sizes, int n_in,
                              void* d_out, int out_size, void* d_ws, size_t ws_size,
                              hipStream_t stream) {
  const float* x      = (const float*)d_in[0];
  const int*   seqlen = (const int*)d_in[1];
  const int*   labels = (const int*)d_in[2];
  const float* W_ih   = (const float*)d_in[3];
  const float* W_hh   = (const float*)d_in[4];
  const float* b_ih   = (const float*)d_in[5];
  const float* b_hh   = (const float*)d_in[6];
  const float* phi_w  = (const float*)d_in[7];
  const float* phi_b  = (const float*)d_in[8];
  const float* psi_w  = (const float*)d_in[9];
  // d_in[10]=psi_b: constant per-b score offset -> cancels in softmax
  const float* cd_w   = (const float*)d_in[11];
  const float* cd_b   = (const float*)d_in[12];
  float* out = (float*)d_out;

  float* w = (float*)d_ws;
  float* ch    = w; w += Bsz * 1024;        // [ctx | h]
  float* c     = w; w += Bsz * Dd;
  float* gates = w; w += Bsz * 2048;
  float* srep  = w; w += Bsz * Hh;
  float* q     = w; w += Bsz * Dd;
  float* Wcat  = w; w += 2048 * 1024;
  float* bb    = w; w += 2048;
  float* psiT  = w; w += Dd * Hh;
  float* pm    = w; w += Bsz * NP;
  float* pl    = w; w += Bsz * NP;
  float* pctx  = w; w += Bsz * NP * Dd;     // ~11.6 MB total

  prep_wcat<<<(2048 * 1024) / 256, 256, 0, stream>>>(W_ih, W_hh, b_ih, b_hh, Wcat, bb);
  prep_psiT<<<(Dd * Hh) / 256, 256, 0, stream>>>(psi_w, psiT);
  init_state<<<(Bsz * Dd) / 256, 256, 0, stream>>>(x, ch, c);

  size_t attn_lds = (size_t)(CHUNK * Dd + CHUNK + 16) * sizeof(float);

  for (int l = 0; l < Ll; ++l) {
    // gates = [ctx|h] @ Wcat^T          (64 x 2048, K=1024) -> 512 wmma tiles
    gemm_wmma_f32<<<128, 128, 0, stream>>>(ch, 1024, Wcat, 1024, nullptr,
                                           gates, 2048, 1024, 128);
    lstm_cell<<<(Bsz * Dd) / 256, 256, 0, stream>>>(gates, bb, W_ih, labels, ch, c, l);
    // s_repr = h @ phi_w^T + phi_b      (64 x 256, K=512)
    gemm_wmma_f32<<<16, 128, 0, stream>>>(ch + 512, 1024, phi_w, Dd, phi_b,
                                          srep, Hh, Dd, Hh / 16);
    // q = s_repr @ psi_w  (via psiT)    (64 x 512, K=256)
    gemm_wmma_f32<<<32, 128, 0, stream>>>(srep, Hh, psiT, Hh, nullptr,
                                          q, Dd, Hh, Dd / 16);
    // fused flash attention over x (TDM-staged LDS tiles)
    attn_partial<<<Bsz * NP, 256, attn_lds, stream>>>(x, q, seqlen, pm, pl, pctx);
    attn_combine<<<Bsz, 256, 0, stream>>>(pm, pl, pctx, ch);
    // raw = [ctx|h] @ cd_w^T + cd_b     (64 x 32, K=1024) -> out[:, l, :]
    gemm_wmma_f32<<<2, 128, 0, stream>>>(ch, 1024, cd_w, 1024, cd_b,
                                         out + l * Vv, Ll * Vv, 1024, Vv / 16);
  }
}